// LayerKVCache_54924041781647
// MI455X (gfx1250) — compile-verified
//
#include <hip/hip_runtime.h>
#include <hip/hip_bf16.h>
#include <stdint.h>

// ---------------- static problem configuration (from reference) ----------------
#define B_      2
#define H_      16
#define L_      16384
#define Dh_     128
#define TPF_    512
#define PD_     1
#define CAP_    (L_ + TPF_)          // 16896
#define BS_     128
#define NBUCK_  32                   // L / TPF / PD
#define KVB_    (CAP_ / BS_)         // 132 kv blocks (exact, no padding)
#define QB_     4                    // ceil(TPF/BS)

#define BH_         (B_ * H_)                    // 32
#define ROWS_PER_T  (BH_ * CAP_)                 // 540672 rows per k-or-v tensor
#define ROWS_TOTAL  (2 * ROWS_PER_T)             // 1081344
#define KV_ELEMS    ((size_t)ROWS_TOTAL * Dh_)   // 138412032 floats

// output layout (floats), concatenated in reference return order
#define OFF_KVNUM   ((size_t)KV_ELEMS)                 // 4
#define OFF_KVIDX   (OFF_KVNUM + 4)                    // 4*132
#define OFF_FULLNUM (OFF_KVIDX + 4 * KVB_)             // 4
#define OFF_FULLIDX (OFF_FULLNUM + 4)                  // 4*132
#define OFF_MASK    (OFF_FULLIDX + 4 * KVB_)           // 16896
#define OFF_WRIT    (OFF_MASK + CAP_)                  // 16896

typedef __attribute__((ext_vector_type(4))) float v4f;
typedef __attribute__((ext_vector_type(4))) int   v4i;
typedef __attribute__((address_space(1))) v4i as1_v4i;   // global (prints as __device__)
typedef __attribute__((address_space(3))) v4i as3_v4i;   // LDS   (prints as __shared__)

// ---------------- CDNA5 async global->LDS helpers ----------------
__device__ __forceinline__ void async_g2l_b128(const void* g, void* lds_generic) {
#if __has_builtin(__builtin_amdgcn_global_load_async_to_lds_b128)
  __builtin_amdgcn_global_load_async_to_lds_b128(
      (as1_v4i*)(unsigned long long)(uintptr_t)g,
      (as3_v4i*)(unsigned)(uintptr_t)lds_generic,
      0, 0);
#else
  unsigned lo = (unsigned)(uintptr_t)lds_generic;        // low 32 bits == LDS byte addr
  unsigned long long ga = (unsigned long long)(uintptr_t)g;
  asm volatile("global_load_async_to_lds_b128 %0, %1, off"
               :: "v"(lo), "v"(ga) : "memory");
#endif
}

__device__ __forceinline__ void wait_async0() {
#if __has_builtin(__builtin_amdgcn_s_wait_asynccnt)
  __builtin_amdgcn_s_wait_asynccnt(0);
#else
  asm volatile("s_wait_asynccnt 0" ::: "memory");
#endif
}

// ---------------- bulk copy: out[0 .. 2*B*H*CAP*Dh) = buf (with substitutions) -
// One float4 per thread; 32 consecutive threads (one wave) cover one 128-float
// row, so the source-select branch is wave-uniform. Non-temporal: 1.1 GB
// streamed exactly once, keep it out of L2's way.
__global__ __launch_bounds__(256) void LayerKVCache_copy_kernel(
    const float* __restrict__ kv, const float* __restrict__ kv_buf,
    const int* __restrict__ t_pos, const int* __restrict__ is_frozen,
    float* __restrict__ out)
{
  const int frame_t = t_pos[0];
  const int frz     = is_frozen[0];
  const int bucket  = (frame_t + (PD_ - 1)) / PD_;
  int slot = bucket % NBUCK_; if (slot < 0) slot += NBUCK_;
  const int rs = slot * TPF_;
  const bool ring_sub = ((frame_t % PD_) == 0) && (frz == 0);

  const unsigned g = blockIdx.x * 256u + threadIdx.x;     // float4 unit id
  if (g >= (unsigned)(ROWS_TOTAL * (Dh_ / 4))) return;
  const int row = (int)(g >> 5);                          // Dh_/4 == 32 units/row
  const int c4  = (int)(g & 31u);

  const int kvsel = row / ROWS_PER_T;                     // 0 = K, 1 = V
  const int rem   = row - kvsel * ROWS_PER_T;
  const int bh    = rem / CAP_;
  const int c     = rem - bh * CAP_;                      // row within CAP

  const float* src;
  if (c >= L_) {
    // current-frame tail slots always take kv
    src = kv + ((size_t)(kvsel * BH_ + bh) * TPF_ + (c - L_)) * Dh_;
  } else if (ring_sub && c >= rs && c < rs + TPF_) {
    // ring-buffer slots take kv when write_step && !frozen
    src = kv + ((size_t)(kvsel * BH_ + bh) * TPF_ + (c - rs)) * Dh_;
  } else {
    src = kv_buf + (size_t)row * Dh_;
  }

  const v4f val = __builtin_nontemporal_load((const v4f*)src + c4);
  __builtin_nontemporal_store(val, (v4f*)out + (size_t)row * (Dh_ / 4) + c4);
}

// ---------------- metadata: mask/written arrays + block-mask ordering ----------
__global__ __launch_bounds__(256) void LayerKVCache_meta_kernel(
    const int* __restrict__ t_pos, const unsigned char* __restrict__ written,
    const int* __restrict__ is_frozen, float* __restrict__ out)
{
  __shared__ unsigned char wlds[CAP_];       // written -> then reused as mask
  __shared__ unsigned char partialf[KVB_];
  __shared__ unsigned char fullf[KVB_];

  const int tid = threadIdx.x;

  // async DMA the 16.9 KB written[] array into LDS (gfx1250 async path)
  for (int u = tid; u < CAP_ / 16; u += 256)
    async_g2l_b128(written + u * 16, &wlds[u * 16]);
  wait_async0();
  __syncthreads();

  const int frame_t = t_pos[0];
  const int frz     = is_frozen[0];
  const int bucket  = (frame_t + (PD_ - 1)) / PD_;
  int slot = bucket % NBUCK_; if (slot < 0) slot += NBUCK_;
  const int rs = slot * TPF_;
  const bool wstep = (frame_t % PD_) == 0;

  // mask_written / updated written, elementwise
  for (int i = tid; i < CAP_; i += 256) {
    const bool w       = wlds[i] != 0;
    const bool in_ring = (i >= rs) && (i < rs + TPF_);
    const bool in_cur  = (i >= L_);
    const bool m  = w && !(wstep && in_ring);
    bool wo = w;
    if (!frz && (wstep ? in_ring : in_cur)) wo = true;
    out[OFF_MASK + i] = m ? 1.0f : 0.0f;
    out[OFF_WRIT + i] = wo ? 1.0f : 0.0f;
    wlds[i] = m ? 1 : 0;                     // keep mask in LDS for reductions
  }
  __syncthreads();

  // per-128-slot block any/all over mask_written
  if (tid < KVB_) {
    bool anyb = false, allb = true;
    const int base = tid * BS_;
    for (int j = 0; j < BS_; ++j) {
      const bool m = wlds[base + j] != 0;
      anyb |= m; allb &= m;
    }
    partialf[tid] = (anyb && !allb) ? 1 : 0;
    fullf[tid]    = allb ? 1 : 0;
  }
  __syncthreads();

  // stable descending argsort of a bool row == flagged indices ascending,
  // then unflagged ascending. 132 entries, replicated over 4 q-blocks.
  if (tid == 0) {
    int np = 0;
    for (int j = 0; j < KVB_; ++j) if (partialf[j]) {
      for (int q = 0; q < QB_; ++q) out[OFF_KVIDX + (size_t)q * KVB_ + np] = (float)j;
      ++np;
    }
    int t = np;
    for (int j = 0; j < KVB_; ++j) if (!partialf[j]) {
      for (int q = 0; q < QB_; ++q) out[OFF_KVIDX + (size_t)q * KVB_ + t] = (float)j;
      ++t;
    }
    int nf = 0;
    for (int j = 0; j < KVB_; ++j) if (fullf[j]) {
      for (int q = 0; q < QB_; ++q) out[OFF_FULLIDX + (size_t)q * KVB_ + nf] = (float)j;
      ++nf;
    }
    t = nf;
    for (int j = 0; j < KVB_; ++j) if (!fullf[j]) {
      for (int q = 0; q < QB_; ++q) out[OFF_FULLIDX + (size_t)q * KVB_ + t] = (float)j;
      ++t;
    }
    for (int q = 0; q < QB_; ++q) {
      out[OFF_KVNUM + q]   = (float)np;
      out[OFF_FULLNUM + q] = (float)nf;
    }
  }
}

extern "C" void kernel_launch(void* const* d_in, const int* in_sizes, int n_in,
                              void* d_out, int out_size, void* d_ws, size_t ws_size,
                              hipStream_t stream) {
  (void)in_sizes; (void)n_in; (void)d_ws; (void)ws_size; (void)out_size;
  const float*         kv        = (const float*)d_in[0];   // (2,B,H,TPF,Dh) f32
  const float*         kv_buf    = (const float*)d_in[1];   // (2,B,H,CAP,Dh) f32
  const int*           t_pos     = (const int*)d_in[2];     // (B,TPF) i32
  const unsigned char* written   = (const unsigned char*)d_in[3]; // (CAP,) bool
  const int*           is_frozen = (const int*)d_in[4];     // scalar
  float* out = (float*)d_out;

  const unsigned units  = (unsigned)(ROWS_TOTAL * (Dh_ / 4)); // 34,603,008
  const unsigned blocks = (units + 255u) / 256u;              // 135,168 (exact)
  LayerKVCache_copy_kernel<<<blocks, 256, 0, stream>>>(kv, kv_buf, t_pos, is_frozen, out);
  LayerKVCache_meta_kernel<<<1, 256, 0, stream>>>(t_pos, written, is_frozen, out);
}